// SharedExpertMOE_4930622456430
// MI455X (gfx1250) — compile-verified
//
#include <hip/hip_runtime.h>
#include <hip/hip_bf16.h>

// ---------------------------------------------------------------------------
// SharedExpertMOE on gfx1250 (CDNA5, wave32), fp32 via V_WMMA_F32_16X16X4_F32
// 2x2 register-blocked WMMA tiles: each wave owns a 32x32 C tile (4 x v8f).
// ---------------------------------------------------------------------------

typedef float v2f __attribute__((ext_vector_type(2)));
typedef float v8f __attribute__((ext_vector_type(8)));

constexpr int Tt   = 8192;   // tokens = B*S
constexpr int Hd   = 1024;   // hidden
constexpr int Ee   = 8;      // experts
constexpr int KTOP = 2;      // top_k
constexpr int NSh  = 2;      // shared experts

// GEMM tiling
constexpr int BM  = 64;      // block tile M (2 wave-rows * 32)
constexpr int BN  = 128;     // block tile N (4 wave-cols * 32)
constexpr int BK  = 32;      // K chunk
constexpr int LDA = 36;      // A row stride in LDS (16B-aligned, conflict-free)
constexpr int LDB = 136;     // B row stride in LDS (16B-aligned, conflict-free)

__device__ __forceinline__ v8f wmma_f32(v2f a, v2f b, v8f c) {
  return __builtin_amdgcn_wmma_f32_16x16x4_f32(false, a, false, b,
                                               (short)0, c, false, false);
}

// ---------------------------------------------------------------------------
// Kernel 1: router. One wave (32 lanes) per token.
// ---------------------------------------------------------------------------
__global__ void moe_router_kernel(const float* __restrict__ x,
                                  const float* __restrict__ gate_w,
                                  const float* __restrict__ gate_b,
                                  float* __restrict__ logits_out,
                                  int*   __restrict__ topk_idx,
                                  float* __restrict__ topk_w) {
  const int wave = threadIdx.x >> 5;
  const int lane = threadIdx.x & 31;
  const int t = blockIdx.x * 8 + wave;   // 8 waves per 256-thread block

  float acc[Ee];
#pragma unroll
  for (int e = 0; e < Ee; ++e) acc[e] = 0.0f;

  const float* xr = x + (size_t)t * Hd;
  for (int h = lane; h < Hd; h += 32) {
    const float xv = xr[h];
    const float* gw = gate_w + h * Ee;   // [H, E] row-major
#pragma unroll
    for (int e = 0; e < Ee; ++e) acc[e] += xv * gw[e];
  }
#pragma unroll
  for (int off = 16; off > 0; off >>= 1) {
#pragma unroll
    for (int e = 0; e < Ee; ++e) acc[e] += __shfl_xor(acc[e], off, 32);
  }

  if (lane == 0) {
    float lg[Ee];
    float m = -3.4e38f;
#pragma unroll
    for (int e = 0; e < Ee; ++e) {
      lg[e] = acc[e] + gate_b[e];
      logits_out[t * Ee + e] = lg[e];
      m = fmaxf(m, lg[e]);
    }
    float p[Ee];
#pragma unroll
    for (int e = 0; e < Ee; ++e) p[e] = __expf(lg[e] - m);
    int i0 = 0; float p0 = p[0];
#pragma unroll
    for (int e = 1; e < Ee; ++e) if (p[e] > p0) { p0 = p[e]; i0 = e; }
    int i1 = -1; float p1 = -1.0f;
#pragma unroll
    for (int e = 0; e < Ee; ++e) if (e != i0 && p[e] > p1) { p1 = p[e]; i1 = e; }
    const float inv = 1.0f / (p0 + p1);   // softmax denom cancels in renorm
    topk_idx[t * KTOP + 0] = i0;
    topk_idx[t * KTOP + 1] = i1;
    topk_w[t * KTOP + 0] = p0 * inv;
    topk_w[t * KTOP + 1] = p1 * inv;
  }
}

// ---------------------------------------------------------------------------
// Kernel 2: deterministic per-(pass k, expert e) compacted token lists.
// ---------------------------------------------------------------------------
__global__ void moe_build_lists_kernel(const int* __restrict__ topk_idx,
                                       int* __restrict__ lists,
                                       int* __restrict__ counts) {
  const int slot = blockIdx.x;            // 0 .. 2*E-1
  const int kp = slot / Ee;
  const int ex = slot % Ee;
  const int lane = threadIdx.x;           // block = 32 threads (one wave)
  int* mylist = lists + (size_t)slot * Tt;

  int cnt = 0;
  for (int base = 0; base < Tt; base += 32) {
    const int t = base + lane;
    const bool match = (topk_idx[t * KTOP + kp] == ex);
    const unsigned mask = (unsigned)__ballot(match);
    if (match) {
      const int pos = __popc(mask & ((1u << lane) - 1u));
      mylist[cnt + pos] = t;
    }
    cnt += __popc(mask);
  }
  if (lane == 0) counts[slot] = cnt;
}

// ---------------------------------------------------------------------------
// Kernel 3: shared experts:  out[t] = sum_s x[t] @ shared_w[s] + sum_s b[s]
// 256 threads = 8 waves in 2x4 grid; wave tile 32x32 (4 accumulators).
// ---------------------------------------------------------------------------
__global__ void moe_shared_gemm_kernel(const float* __restrict__ x,
                                       const float* __restrict__ shared_w,
                                       const float* __restrict__ shared_b,
                                       float* __restrict__ out) {
  __shared__ __align__(16) float lA[BM * LDA];
  __shared__ __align__(16) float lB[BK * LDB];

  const int tid = threadIdx.x;
  const int wv = tid >> 5, lane = tid & 31;
  const int half = lane >> 4, lm = lane & 15;
  const int K0 = half * 2;
  const int mbase = (wv >> 2) * 32;        // wave-row: 0 or 32
  const int nbase = (wv & 3) * 32;         // wave-col: 0,32,64,96
  const int n0 = blockIdx.x * BN;
  const int t0 = blockIdx.y * BM;

  v8f c00 = {}, c01 = {}, c10 = {}, c11 = {};

  for (int kb = 0; kb < Hd; kb += BK) {
    // stage A tile [BM][BK] as float4 (tokens are contiguous rows)
#pragma unroll
    for (int i = tid; i < BM * (BK / 4); i += 256) {
      const int r = i >> 3, q = i & 7;
      const float4 v = *(const float4*)&x[(size_t)(t0 + r) * Hd + kb + q * 4];
      *(float4*)&lA[r * LDA + q * 4] = v;
    }
    for (int s = 0; s < NSh; ++s) {
      // stage B tile [BK][BN] as float4 from shared_w[s]
#pragma unroll
      for (int i = tid; i < BK * (BN / 4); i += 256) {
        const int r = i >> 5, q = i & 31;
        const float4 v =
            *(const float4*)&shared_w[(size_t)(s * Hd + kb + r) * Hd + n0 + q * 4];
        *(float4*)&lB[r * LDB + q * 4] = v;
      }
      __syncthreads();
#pragma unroll
      for (int kk = 0; kk < BK; kk += 4) {
        v2f a0, a1, b0, b1;
        a0.x = lA[(mbase + lm) * LDA + kk + K0];
        a0.y = lA[(mbase + lm) * LDA + kk + K0 + 1];
        a1.x = lA[(mbase + 16 + lm) * LDA + kk + K0];
        a1.y = lA[(mbase + 16 + lm) * LDA + kk + K0 + 1];
        b0.x = lB[(kk + K0) * LDB + nbase + lm];
        b0.y = lB[(kk + K0 + 1) * LDB + nbase + lm];
        b1.x = lB[(kk + K0) * LDB + nbase + 16 + lm];
        b1.y = lB[(kk + K0 + 1) * LDB + nbase + 16 + lm];
        c00 = wmma_f32(a0, b0, c00);
        c01 = wmma_f32(a0, b1, c01);
        c10 = wmma_f32(a1, b0, c10);
        c11 = wmma_f32(a1, b1, c11);
      }
      __syncthreads();
    }
  }

  const int col0 = n0 + nbase + lm;
  const int col1 = col0 + 16;
  const float sb0 = shared_b[col0] + shared_b[Hd + col0];
  const float sb1 = shared_b[col1] + shared_b[Hd + col1];
#pragma unroll
  for (int j = 0; j < 8; ++j) {
    const int m0r = t0 + mbase + j + half * 8;       // rows of c0x
    const int m1r = m0r + 16;                        // rows of c1x
    out[(size_t)m0r * Hd + col0] = c00[j] + sb0;
    out[(size_t)m0r * Hd + col1] = c01[j] + sb1;
    out[(size_t)m1r * Hd + col0] = c10[j] + sb0;
    out[(size_t)m1r * Hd + col1] = c11[j] + sb1;
  }
}

// ---------------------------------------------------------------------------
// Kernel 4: routed experts, one pass per top-k slot (kpass = 0 or 1).
// Gathered A rows; weight-scaled accumulate into out (no atomics: each token
// appears exactly once per pass; passes are stream-serialized).
// ---------------------------------------------------------------------------
__global__ void moe_expert_gemm_kernel(const float* __restrict__ x,
                                       const float* __restrict__ expert_w,
                                       const float* __restrict__ expert_b,
                                       const int* __restrict__ lists,
                                       const int* __restrict__ counts,
                                       const float* __restrict__ topk_w,
                                       float* __restrict__ out,
                                       int kpass) {
  const int ex = blockIdx.z;
  const int slot = kpass * Ee + ex;
  const int cnt = counts[slot];
  const int m0 = blockIdx.y * BM;
  if (m0 >= cnt) return;

  __shared__ __align__(16) float lA[BM * LDA];
  __shared__ __align__(16) float lB[BK * LDB];
  __shared__ int   tokRow[BM];
  __shared__ float wRow[BM];

  const int tid = threadIdx.x;
  const int wv = tid >> 5, lane = tid & 31;
  const int half = lane >> 4, lm = lane & 15;
  const int K0 = half * 2;
  const int mbase = (wv >> 2) * 32;
  const int nbase = (wv & 3) * 32;
  const int n0 = blockIdx.x * BN;
  const int* mylist = lists + (size_t)slot * Tt;

  if (tid < BM) {
    const int mi = m0 + tid;
    if (mi < cnt) {
      const int t = mylist[mi];
      tokRow[tid] = t;
      wRow[tid] = topk_w[t * KTOP + kpass];
    } else {
      tokRow[tid] = mylist[m0];   // safe address for loads; store is masked
      wRow[tid] = 0.0f;
    }
  }
  __syncthreads();

  v8f c00 = {}, c01 = {}, c10 = {}, c11 = {};
  const float* We = expert_w + (size_t)ex * Hd * Hd;

  for (int kb = 0; kb < Hd; kb += BK) {
#pragma unroll
    for (int i = tid; i < BM * (BK / 4); i += 256) {
      const int r = i >> 3, q = i & 7;
      const float4 v = *(const float4*)&x[(size_t)tokRow[r] * Hd + kb + q * 4];
      *(float4*)&lA[r * LDA + q * 4] = v;
    }
#pragma unroll
    for (int i = tid; i < BK * (BN / 4); i += 256) {
      const int r = i >> 5, q = i & 31;
      const float4 v = *(const float4*)&We[(size_t)(kb + r) * Hd + n0 + q * 4];
      *(float4*)&lB[r * LDB + q * 4] = v;
    }
    __syncthreads();
#pragma unroll
    for (int kk = 0; kk < BK; kk += 4) {
      v2f a0, a1, b0, b1;
      a0.x = lA[(mbase + lm) * LDA + kk + K0];
      a0.y = lA[(mbase + lm) * LDA + kk + K0 + 1];
      a1.x = lA[(mbase + 16 + lm) * LDA + kk + K0];
      a1.y = lA[(mbase + 16 + lm) * LDA + kk + K0 + 1];
      b0.x = lB[(kk + K0) * LDB + nbase + lm];
      b0.y = lB[(kk + K0 + 1) * LDB + nbase + lm];
      b1.x = lB[(kk + K0) * LDB + nbase + 16 + lm];
      b1.y = lB[(kk + K0 + 1) * LDB + nbase + 16 + lm];
      c00 = wmma_f32(a0, b0, c00);
      c01 = wmma_f32(a0, b1, c01);
      c10 = wmma_f32(a1, b0, c10);
      c11 = wmma_f32(a1, b1, c11);
    }
    __syncthreads();
  }

  const int col0 = n0 + nbase + lm;
  const int col1 = col0 + 16;
  const float bias0 = expert_b[ex * Hd + col0];
  const float bias1 = expert_b[ex * Hd + col1];
#pragma unroll
  for (int j = 0; j < 8; ++j) {
    const int lm0 = mbase + j + half * 8;   // local row of c0x
    const int lm1 = lm0 + 16;               // local row of c1x
    if (m0 + lm0 < cnt) {
      const int t = tokRow[lm0];
      const float w = wRow[lm0];
      out[(size_t)t * Hd + col0] += w * (c00[j] + bias0);
      out[(size_t)t * Hd + col1] += w * (c01[j] + bias1);
    }
    if (m0 + lm1 < cnt) {
      const int t = tokRow[lm1];
      const float w = wRow[lm1];
      out[(size_t)t * Hd + col0] += w * (c10[j] + bias0);
      out[(size_t)t * Hd + col1] += w * (c11[j] + bias1);
    }
  }
}

// ---------------------------------------------------------------------------
extern "C" void kernel_launch(void* const* d_in, const int* in_sizes, int n_in,
                              void* d_out, int out_size, void* d_ws, size_t ws_size,
                              hipStream_t stream) {
  const float* x        = (const float*)d_in[0];
  const float* gate_w   = (const float*)d_in[1];
  const float* gate_b   = (const float*)d_in[2];
  const float* expert_w = (const float*)d_in[3];
  const float* expert_b = (const float*)d_in[4];
  const float* shared_w = (const float*)d_in[5];
  const float* shared_b = (const float*)d_in[6];
  (void)in_sizes; (void)n_in; (void)out_size; (void)ws_size;

  float* out        = (float*)d_out;
  float* logits_out = out + (size_t)Tt * Hd;   // second output segment

  // workspace layout (~0.7 MB)
  char* ws = (char*)d_ws;
  int*   topk_idx = (int*)ws;                                  // T*K ints
  float* topk_w   = (float*)(ws + (size_t)Tt * KTOP * 4);      // T*K floats
  int*   counts   = (int*)(ws + (size_t)2 * Tt * KTOP * 4);    // 2*E ints
  int*   lists    = counts + 64;                               // 2*E*T ints

  // 1) router
  moe_router_kernel<<<dim3(Tt / 8), dim3(256), 0, stream>>>(
      x, gate_w, gate_b, logits_out, topk_idx, topk_w);

  // 2) deterministic list build
  moe_build_lists_kernel<<<dim3(KTOP * Ee), dim3(32), 0, stream>>>(
      topk_idx, lists, counts);

  // 3) shared experts -> initializes out
  moe_shared_gemm_kernel<<<dim3(Hd / BN, Tt / BM), dim3(256), 0, stream>>>(
      x, shared_w, shared_b, out);

  // 4) routed experts, stream-serialized passes => deterministic, race-free
  for (int kpass = 0; kpass < KTOP; ++kpass) {
    moe_expert_gemm_kernel<<<dim3(Hd / BN, (Tt + BM - 1) / BM, Ee),
                             dim3(256), 0, stream>>>(
        x, expert_w, expert_b, lists, counts, topk_w, out, kpass);
  }
}